// IndividualStacksLayer_24790551232734
// MI455X (gfx1250) — compile-verified
//
#include <hip/hip_runtime.h>

#define BLK 256
constexpr int HH = 512, WW = 512, CC = 2;
constexpr int ROWF  = WW * CC;   // 1024 floats per image row
constexpr int ROWF4 = ROWF / 4;  // 256 float4 per image row

#define LDS_AS __attribute__((address_space(3)))
#define GLB_AS __attribute__((address_space(1)))

typedef int v4i __attribute__((vector_size(4 * sizeof(int))));

#if __has_builtin(__builtin_amdgcn_global_load_async_to_lds_b128)
#define HAVE_ASYNC 1
#else
#define HAVE_ASYNC 0
#endif

__device__ __forceinline__ void wait_async0() {
#if __has_builtin(__builtin_amdgcn_s_wait_asynccnt)
  __builtin_amdgcn_s_wait_asynccnt(0);
#else
  asm volatile("s_wait_asynccnt 0" ::: "memory");
#endif
}

// One block per (global output image, output row). Each thread produces 2
// pixels (= one float4 of N,H,W,C fp32 output). The two source rows (y0c,
// y1c after clamp) are staged into LDS via CDNA5 async global->LDS copies.
__global__ __launch_bounds__(BLK) void stn_shift_rows(
    const float* __restrict__ img0, const float* __restrict__ tr0, const float* __restrict__ st0,
    const float* __restrict__ img1, const float* __restrict__ tr1, const float* __restrict__ st1,
    const int* __restrict__ pb0p, const int* __restrict__ pa0p,
    const int* __restrict__ pb1p, const int* __restrict__ pa1p,
    int nz0, int nz1, float* __restrict__ out)
{
  __shared__ float lds[2 * ROWF];  // 8 KB: two staged input rows

  const int row  = blockIdx.x;   // 0..H-1
  const int gimg = blockIdx.z;   // 0..(F0+F1-1) over the concatenated output
  const int t    = threadIdx.x;  // 0..255

  const int pb0 = *pb0p, pa0 = *pa0p, pb1 = *pb1p;
  (void)pa1p;
  const int F0 = pb0 + nz0 + pa0;

  const float* img; const float* tr; const float* st;
  int local, nz;
  if (gimg < F0) { img = img0; tr = tr0; st = st0; local = gimg - pb0;       nz = nz0; }
  else           { img = img1; tr = tr1; st = st1; local = gimg - F0 - pb1;  nz = nz1; }

  float4* orow4 = reinterpret_cast<float4*>(out + ((size_t)gimg * HH + row) * ROWF);

  if (local < 0 || local >= nz) {           // zero-padding image: stream zeros
    orow4[t] = make_float4(0.f, 0.f, 0.f, 0.f);
    return;
  }

  // ----- per-image translation, matching the reference math exactly -----
  const float ax  = tr[2 * local]     + st[0];
  const float ay  = tr[2 * local + 1] + st[1];
  const float ntx = ax / (HH * 0.5f);       // norm_t[:,0] = abs_t0 / (H/2)
  const float nty = ay / (WW * 0.5f);       // norm_t[:,1] = abs_t1 / (W/2)

  // y = 0.5*(y_t + nty + 1)*(H-1),  y_t = linspace(-1,1,H)[row]
  const float y_t = fmaf((float)row, 2.0f / (HH - 1), -1.0f);
  const float y   = 0.5f * (y_t + nty + 1.0f) * (HH - 1);
  const int   y0i = (int)floorf(y);
  const int   y0c = min(max(y0i,     0), HH - 1);
  const int   y1c = min(max(y0i + 1, 0), HH - 1);
  const float wy1 = (float)y1c - y;         // (y1f - y)  [clip-gather weights]
  const float wy0 = y - (float)y0c;         // (y - y0f)

  const float* grow0 = img + ((size_t)local * HH + y0c) * ROWF;
  const float* grow1 = img + ((size_t)local * HH + y1c) * ROWF;

  // ----- stage the two source rows into LDS (async/ASYNCcnt path) -----
#if HAVE_ASYNC
  {
    const float* g0 = grow0 + t * 4;        // 16 B per lane per row
    const float* g1 = grow1 + t * 4;
    float* l0 = &lds[t * 4];
    float* l1 = &lds[ROWF + t * 4];
    __builtin_amdgcn_global_load_async_to_lds_b128((GLB_AS v4i*)g0, (LDS_AS v4i*)l0, 0, 0);
    __builtin_amdgcn_global_load_async_to_lds_b128((GLB_AS v4i*)g1, (LDS_AS v4i*)l1, 0, 0);
    wait_async0();
  }
#else
  reinterpret_cast<float4*>(lds)[t]         = reinterpret_cast<const float4*>(grow0)[t];
  reinterpret_cast<float4*>(lds)[ROWF4 + t] = reinterpret_cast<const float4*>(grow1)[t];
#endif
  __syncthreads();

  const float2* r0 = reinterpret_cast<const float2*>(&lds[0]);      // row y0c
  const float2* r1 = reinterpret_cast<const float2*>(&lds[ROWF]);   // row y1c

  float o[4];
#pragma unroll
  for (int k = 0; k < 2; ++k) {
    const int j = 2 * t + k;
    const float x_t = fmaf((float)j, 2.0f / (WW - 1), -1.0f);
    const float x   = 0.5f * (x_t + ntx + 1.0f) * (WW - 1);
    const int   x0i = (int)floorf(x);
    const int   x0c = min(max(x0i,     0), WW - 1);
    const int   x1c = min(max(x0i + 1, 0), WW - 1);
    const float wx1 = (float)x1c - x;       // (x1f - x)
    const float wx0 = x - (float)x0c;       // (x - x0f)
    const float wa = wx1 * wy1;             // * img[y0c, x0c]
    const float wb = wx1 * wy0;             // * img[y1c, x0c]
    const float wc = wx0 * wy1;             // * img[y0c, x1c]
    const float wd = wx0 * wy0;             // * img[y1c, x1c]
    const float2 Ia = r0[x0c], Ib = r1[x0c], Ic = r0[x1c], Id = r1[x1c];
    o[2 * k + 0] = wa * Ia.x + wb * Ib.x + wc * Ic.x + wd * Id.x;
    o[2 * k + 1] = wa * Ia.y + wb * Ib.y + wc * Ic.y + wd * Id.y;
  }
  orow4[t] = make_float4(o[0], o[1], o[2], o[3]);
}

extern "C" void kernel_launch(void* const* d_in, const int* in_sizes, int n_in,
                              void* d_out, int out_size, void* d_ws, size_t ws_size,
                              hipStream_t stream) {
  (void)n_in; (void)d_ws; (void)ws_size;
  const float* img0 = (const float*)d_in[0];
  const float* tr0  = (const float*)d_in[1];
  const float* st0  = (const float*)d_in[2];
  const float* img1 = (const float*)d_in[3];
  const float* tr1  = (const float*)d_in[4];
  const float* st1  = (const float*)d_in[5];
  const int*   pb0  = (const int*)d_in[6];
  const int*   pa0  = (const int*)d_in[7];
  const int*   pb1  = (const int*)d_in[8];
  const int*   pa1  = (const int*)d_in[9];

  const int nz0   = in_sizes[0] / (HH * WW * CC);  // 52
  const int nz1   = in_sizes[3] / (HH * WW * CC);  // 48
  const int total = out_size    / (HH * WW * CC);  // 128 padded output images

  dim3 grid(HH, 1, total);   // one block per output row per image
  stn_shift_rows<<<grid, BLK, 0, stream>>>(img0, tr0, st0, img1, tr1, st1,
                                           pb0, pa0, pb1, pa1, nz0, nz1,
                                           (float*)d_out);
}